// ScatterLoss_15264313770529
// MI455X (gfx1250) — compile-verified
//
#include <hip/hip_runtime.h>
#include <hip/hip_bf16.h>

typedef __attribute__((ext_vector_type(2))) float v2f;
typedef __attribute__((ext_vector_type(8))) float v8f;

#define N_ROWS   8192
#define DIM      256
#define CLS_PAD  112                      // 100 classes padded to 7 tiles of 16
#define NTILES   7                        // class tiles per wave
#define KSPLIT   32
#define ROWS_PER_SPLIT (N_ROWS / KSPLIT)  // 256 rows per split

// -------- kernel 0: zero the class-sum accumulator S[112][256] --------------
__global__ void zeroS_kernel(float* __restrict__ S) {
    S[blockIdx.x * 256 + threadIdx.x] = 0.0f;
}

// -------- kernel 1: per-row inverse L2 norm (one wave32 per row) ------------
__global__ void rownorm_kernel(const float* __restrict__ e,
                               float* __restrict__ invn) {
    const int wave = threadIdx.x >> 5;
    const int lane = threadIdx.x & 31;
    const int row  = blockIdx.x * 8 + wave;
    const float4* p = (const float4*)(e + (size_t)row * DIM);
    float4 a = p[lane];        // dims [lane*4 .. lane*4+3]
    float4 b = p[lane + 32];   // dims [128+lane*4 ..]
    float s = a.x*a.x + a.y*a.y + a.z*a.z + a.w*a.w
            + b.x*b.x + b.y*b.y + b.z*b.z + b.w*b.w;
    #pragma unroll
    for (int off = 16; off > 0; off >>= 1) s += __shfl_xor(s, off, 32);
    if (lane == 0) invn[row] = 1.0f / sqrtf(s);
}

// -------- kernel 2: S += scaled-one-hot^T @ E via V_WMMA_F32_16X16X4_F32 ----
// One wave owns one 16-dim column tile and ALL 7 class tiles (7 accumulators),
// so each per-lane B load pair feeds 7 WMMAs. A-side data (labels, invnorms)
// is loaded via wave-uniform int4/float4 (scalar loads) and selected
// branchlessly per half-wave.
// Layout (ISA 7.12.2): A VGPR0 = K0 (lanes 0-15) / K2 (lanes 16-31),
//                      A VGPR1 = K1 / K3; B mirrored; C/D: VGPR r -> M=r (+8 hi half).
__global__ void wmma_classsum_kernel(const float* __restrict__ e,
                                     const int*   __restrict__ y,
                                     const float* __restrict__ invn,
                                     float*       __restrict__ S) {
    const int lane    = threadIdx.x;      // wave32, no divergence -> EXEC all 1s
    const int dimTile = blockIdx.x;       // 0..15
    const int split   = blockIdx.y;       // 0..KSPLIT-1
    const int half    = lane >> 4;        // 0: rows k,k+1   1: rows k+2,k+3
    const int m       = lane & 15;
    const int col     = dimTile * 16 + m;

    v8f acc[NTILES];
    #pragma unroll
    for (int t = 0; t < NTILES; ++t) acc[t] = (v8f){0,0,0,0,0,0,0,0};

    const int kBeg = split * ROWS_PER_SPLIT;
    const int kEnd = kBeg + ROWS_PER_SPLIT;

    #pragma unroll 2
    for (int k = kBeg; k < kEnd; k += 4) {
        // Wave-uniform A-side loads (lower to s_load_b128 / KMcnt)
        const int4   ys  = *(const int4*)(y + k);
        const float4 ivs = *(const float4*)(invn + k);
        // Branchless per-half selects (v_cndmask, no EXEC manipulation)
        const int   yx  = half ? ys.z  : ys.x;
        const int   yy  = half ? ys.w  : ys.y;
        const float ivx = half ? ivs.z : ivs.x;
        const float ivy = half ? ivs.w : ivs.y;

        // Per-lane B tile: rows (k+2*half, k+2*half+1), column `col`
        const int kx = k + 2 * half;
        v2f b = { e[(size_t)kx       * DIM + col],
                  e[(size_t)(kx + 1) * DIM + col] };

        #pragma unroll
        for (int t = 0; t < NTILES; ++t) {
            const int cls = t * 16 + m;
            v2f a = { (yx == cls) ? ivx : 0.0f,
                      (yy == cls) ? ivy : 0.0f };
            acc[t] = __builtin_amdgcn_wmma_f32_16x16x4_f32(
                         /*neg_a=*/false, a, /*neg_b=*/false, b,
                         /*c_mod=*/(short)0, acc[t],
                         /*reuse_a=*/false, /*reuse_b=*/false);
        }
    }

    #pragma unroll
    for (int t = 0; t < NTILES; ++t) {
        #pragma unroll
        for (int r = 0; r < 8; ++r) {
            const int cls = t * 16 + r + 8 * half;   // C/D: M = r (+8 for hi half)
            atomicAdd(&S[cls * DIM + col], acc[t][r]);
        }
    }
}

// -------- kernel 3: histogram + final scalar reduction ----------------------
__global__ void finalize_kernel(const int* __restrict__ y,
                                const float* __restrict__ S,
                                float* __restrict__ out) {
    __shared__ int   hist[CLS_PAD];
    __shared__ float redA[256];
    __shared__ float redB[256];
    const int tid = threadIdx.x;

    if (tid < CLS_PAD) hist[tid] = 0;
    __syncthreads();
    for (int i = tid; i < N_ROWS; i += 256) {
        int c = y[i];
        if (c >= 0 && c < CLS_PAD) atomicAdd(&hist[c], 1);
    }
    __syncthreads();

    // per-dim column: total = sum_c S[c][d], ssq = sum_c S[c][d]^2
    float tot = 0.0f, ssq = 0.0f;
    for (int c = 0; c < CLS_PAD; ++c) {
        float v = S[c * DIM + tid];
        tot += v;
        ssq += v * v;
    }
    redA[tid] = tot * tot;   // -> ||sum_c S_c||^2
    redB[tid] = ssq;         // -> sum_c ||S_c||^2
    __syncthreads();

    if (tid == 0) {
        float sumA = 0.0f, sumB = 0.0f;
        for (int i = 0; i < 256; ++i) { sumA += redA[i]; sumB += redB[i]; }
        long long sn2 = 0;
        for (int c = 0; c < CLS_PAD; ++c) {
            long long n = hist[c];
            sn2 += n * n;
        }
        long long cnt2 = (long long)N_ROWS * (long long)N_ROWS - sn2; // = 2*cnter
        out[0] = (sumA - sumB) / (float)cnt2;  // 0.5*(A-B) / cnter
    }
}

extern "C" void kernel_launch(void* const* d_in, const int* in_sizes, int n_in,
                              void* d_out, int out_size, void* d_ws, size_t ws_size,
                              hipStream_t stream) {
    const float* e = (const float*)d_in[0];   // 8192 x 256 f32
    const int*   y = (const int*)d_in[1];     // 8192 labels
    float* out  = (float*)d_out;              // scalar f32

    float* invn = (float*)d_ws;               // 8192 f32   (32 KB)
    float* S    = invn + N_ROWS;              // 112x256 f32 (~115 KB)

    zeroS_kernel<<<CLS_PAD, 256, 0, stream>>>(S);
    rownorm_kernel<<<N_ROWS / 8, 256, 0, stream>>>(e, invn);
    dim3 g2(16, KSPLIT);                      // dim tiles x K-splits
    wmma_classsum_kernel<<<g2, 32, 0, stream>>>(e, y, invn, S);
    finalize_kernel<<<1, 256, 0, stream>>>(y, S, out);
}